// MaxRetrievalModel_7748121002088
// MI455X (gfx1250) — compile-verified
//
#include <hip/hip_runtime.h>

// MaxRetrievalModel on MI455X (gfx1250, wave32).
//
// Algebraic restructuring removes both big GEMMs:
//   scores[b,n] = x[b,n,:] . (Wk q_b) + q_b.bk          (one 256MB stream of x)
//   z_b         = (sum_n attn[b,n] x[b,n,:]) Wv + bv    (sparse second touch)
// leaving a bandwidth-bound GEMV pass + tiny fp32 GEMMs done with
// V_WMMA_F32_16X16X4_F32 wave tiles.

typedef __attribute__((ext_vector_type(2))) float v2f;
typedef __attribute__((ext_vector_type(8))) float v8f;

#define DEMB 512   // D (embedding) == D_in == 512 in this model

__device__ __forceinline__ float wave_reduce_add(float v) {
#pragma unroll
  for (int off = 16; off >= 1; off >>= 1) v += __shfl_xor(v, off, 32);
  return v;
}

// ---------------------------------------------------------------------------
// One wave computes a 16x16 f32 output tile of D = A * B (+bias) using
// V_WMMA_F32_16X16X4_F32, accumulating over K in steps of 4.
// A: [M x K] row-major (lda).  B: [K x N] row-major (ldb), or if TRANSB,
// Bm is [N x K] row-major and we use Bm^T.
// f32 16x4 A layout: lanes 0-15 hold K={0,1} in v[0],v[1]; lanes 16-31 K={2,3}.
// B (4x16) mirrors that with N across lanes. C/D: VGPR r = row (r + 8*half).
// GUARD=false: no column bounds checks (full tiles) -> branch-free k-loop.
// GUARD=true:  clamped address + 0/1 mask multiply   -> still branch-free.
// ---------------------------------------------------------------------------
template <bool TRANSB, bool GUARD>
__device__ __forceinline__ void wmma_tile_f32(
    const float* __restrict__ A, int lda,
    const float* __restrict__ Bm, int ldb,
    const float* __restrict__ bias,
    float* __restrict__ D, int ldd,
    int m0, int n0, int K, int nLimit) {
  const int lane = threadIdx.x & 31;
  const int half = lane >> 4;     // 0: K pair {0,1}, 1: K pair {2,3}
  const int mr   = lane & 15;
  const int nr   = lane & 15;
  const int n    = n0 + nr;

  bool  nok = true;
  int   nc  = n;     // clamped (always in-bounds) column
  float msk = 1.f;   // 0 for out-of-range columns
  if (GUARD) {
    nok = (n < nLimit);
    nc  = nok ? n : (nLimit - 1);
    msk = nok ? 1.f : 0.f;
  }

  v8f c = {0.f, 0.f, 0.f, 0.f, 0.f, 0.f, 0.f, 0.f};
  const float* arow = A + (size_t)(m0 + mr) * lda + 2 * half;
  const float* bcol = TRANSB ? (Bm + (size_t)nc * ldb + 2 * half)
                             : (Bm + (size_t)(2 * half) * ldb + nc);

  for (int k0 = 0; k0 < K; k0 += 4) {
    v2f a, b;
    a[0] = arow[k0];
    a[1] = arow[k0 + 1];
    if (TRANSB) {
      b[0] = bcol[k0];
      b[1] = bcol[k0 + 1];
    } else {
      b[0] = bcol[(size_t)k0 * ldb];
      b[1] = bcol[(size_t)(k0 + 1) * ldb];
    }
    if (GUARD) { b[0] *= msk; b[1] *= msk; }
    c = __builtin_amdgcn_wmma_f32_16x16x4_f32(false, a, false, b, (short)0, c,
                                              false, false);
  }
  if (!GUARD || nok) {
    const float bb = bias ? bias[n] : 0.f;
#pragma unroll
    for (int r = 0; r < 8; ++r) {
      const int m = m0 + r + 8 * half;
      D[(size_t)m * ldd + n] = c[r] + bb;
    }
  }
}

// --- 0: q[b,d] = x_query[b]*Wq[d] + bq[d] -----------------------------------
__global__ void q_kernel(const float* __restrict__ xq,
                         const float* __restrict__ Wq,
                         const float* __restrict__ bq,
                         float* __restrict__ Q, int total) {
  int idx = blockIdx.x * blockDim.x + threadIdx.x;
  if (idx >= total) return;
  int b = idx >> 9, d = idx & (DEMB - 1);
  Q[idx] = xq[b] * Wq[d] + bq[d];
}

// --- 0b: qbk[b] = q_b . bk  (one wave per b, one block) ---------------------
__global__ void qbk_kernel(const float* __restrict__ Q,
                           const float* __restrict__ bk,
                           float* __restrict__ qbk, int B) {
  int wave = threadIdx.x >> 5, lane = threadIdx.x & 31;
  if (wave >= B) return;
  float acc = 0.f;
  for (int d = lane; d < DEMB; d += 32) acc += Q[wave * DEMB + d] * bk[d];
  acc = wave_reduce_add(acc);
  if (lane == 0) qbk[wave] = acc;
}

// --- 1: wk = Q @ Wk^T   (WMMA, TRANSB, no guard: N=512 exact) ---------------
__global__ void gemm_wk_kernel(const float* __restrict__ Q,
                               const float* __restrict__ Wk,
                               float* __restrict__ wk) {
  wmma_tile_f32<true, false>(Q, DEMB, Wk, DEMB, nullptr, wk, DEMB,
                             blockIdx.x * 16, blockIdx.y * 16, DEMB, DEMB);
}

// --- 2: scores[b,n] = (x[b,n,:].wk_b + qbk[b]) * D^-0.5 ---------------------
// Wave-per-row; lane covers 16 contiguous floats (4x b128 loads). This is the
// single bandwidth-bound pass over x_items (256 MB).
__global__ void score_kernel(const float* __restrict__ x,
                             const float* __restrict__ wk,
                             const float* __restrict__ qbk,
                             float* __restrict__ scores, int N) {
  __shared__ float swk[DEMB];
  const int rowsPerBlk = 8;  // 8 waves per 256-thread block
  int blocksPerB = N / rowsPerBlk;
  int b = blockIdx.x / blocksPerB;
  int rowblk = blockIdx.x % blocksPerB;
  for (int i = threadIdx.x; i < DEMB; i += blockDim.x) swk[i] = wk[b * DEMB + i];
  __syncthreads();

  int wave = threadIdx.x >> 5, lane = threadIdx.x & 31;
  int n = rowblk * rowsPerBlk + wave;
  const float4* x4 = (const float4*)(x + ((size_t)b * N + n) * DEMB + lane * 16);
  const float4* w4 = (const float4*)(swk + lane * 16);
  // hint the streaming pattern (global_prefetch_b8): next block, same slot
  __builtin_prefetch(x4 + (size_t)rowsPerBlk * (DEMB / 4), 0, 0);
  float acc = 0.f;
#pragma unroll
  for (int j = 0; j < 4; ++j) {
    float4 xv = x4[j], wv = w4[j];
    acc += xv.x * wv.x + xv.y * wv.y + xv.z * wv.z + xv.w * wv.w;
  }
  acc = wave_reduce_add(acc);
  if (lane == 0)
    scores[(size_t)b * N + n] = (acc + qbk[b]) * 0.044194173824159216f; // 512^-0.5
}

// --- 3: sparsemax tau[b] via bisection on sum(relu(z - tau)) = 1 ------------
// tau is the unique root in [zmax-1, zmax]; scores cached in LDS.
__global__ void tau_kernel(const float* __restrict__ scores,
                           float* __restrict__ tau, int N) {
  __shared__ float z[4096];
  __shared__ float red[256];
  int b = blockIdx.x, t = threadIdx.x;
  float lmax = -3.0e38f;
  for (int i = t; i < N; i += 256) {
    float v = scores[(size_t)b * N + i];
    z[i] = v;
    lmax = fmaxf(lmax, v);
  }
  red[t] = lmax;
  __syncthreads();
  for (int s = 128; s > 0; s >>= 1) {
    if (t < s) red[t] = fmaxf(red[t], red[t + s]);
    __syncthreads();
  }
  float zmax = red[0];
  __syncthreads();

  float lo = zmax - 1.0f, hi = zmax;  // f(lo) >= 0 >= f(hi)
  for (int it = 0; it < 48; ++it) {
    float mid = 0.5f * (lo + hi);
    float ls = 0.f;
    for (int i = t; i < N; i += 256) ls += fmaxf(z[i] - mid, 0.f);
    red[t] = ls;
    __syncthreads();
    for (int s = 128; s > 0; s >>= 1) {
      if (t < s) red[t] += red[t + s];
      __syncthreads();
    }
    float sum = red[0];
    __syncthreads();
    if (sum >= 1.0f) lo = mid; else hi = mid;  // uniform branch
  }
  if (t == 0) tau[b] = 0.5f * (lo + hi);
}

// --- 4: partial u: part[b,chunk,:] = sum_{n in chunk} relu(s-tau) * x[b,n,:]
// Sparsemax support is tiny, so nearly every row is skipped (wave-uniform
// branch => no x traffic for zero-weight rows). No atomics: deterministic.
__global__ void upart_kernel(const float* __restrict__ x,
                             const float* __restrict__ scores,
                             const float* __restrict__ tau,
                             float* __restrict__ part, int N, int nchunks) {
  __shared__ float ush[8 * DEMB];
  int b = blockIdx.y, chunk = blockIdx.x;
  int wave = threadIdx.x >> 5, lane = threadIdx.x & 31;
  float tb = tau[b];
  float4 a0 = {0,0,0,0}, a1 = {0,0,0,0}, a2 = {0,0,0,0}, a3 = {0,0,0,0};
  int base = chunk * 256 + wave * 32;
  for (int r = 0; r < 32; ++r) {
    int n = base + r;
    float a = scores[(size_t)b * N + n] - tb;  // same value across the wave
    if (a > 0.f) {
      const float4* x4 =
          (const float4*)(x + ((size_t)b * N + n) * DEMB + lane * 16);
      float4 v;
      v = x4[0]; a0.x += a*v.x; a0.y += a*v.y; a0.z += a*v.z; a0.w += a*v.w;
      v = x4[1]; a1.x += a*v.x; a1.y += a*v.y; a1.z += a*v.z; a1.w += a*v.w;
      v = x4[2]; a2.x += a*v.x; a2.y += a*v.y; a2.z += a*v.z; a2.w += a*v.w;
      v = x4[3]; a3.x += a*v.x; a3.y += a*v.y; a3.z += a*v.z; a3.w += a*v.w;
    }
  }
  float* dst = ush + wave * DEMB + lane * 16;
  ((float4*)dst)[0] = a0; ((float4*)dst)[1] = a1;
  ((float4*)dst)[2] = a2; ((float4*)dst)[3] = a3;
  __syncthreads();
  for (int i = threadIdx.x; i < DEMB; i += 256) {
    float s = 0.f;
#pragma unroll
    for (int w = 0; w < 8; ++w) s += ush[w * DEMB + i];
    part[((size_t)b * nchunks + chunk) * DEMB + i] = s;
  }
}

// --- 4b: u[b,i] = sum_c part[b,c,i]  (fixed order => deterministic) ---------
__global__ void ureduce_kernel(const float* __restrict__ part,
                               float* __restrict__ u, int nchunks, int total) {
  int idx = blockIdx.x * blockDim.x + threadIdx.x;
  if (idx >= total) return;
  int b = idx >> 9, i = idx & (DEMB - 1);
  float s = 0.f;
  for (int c = 0; c < nchunks; ++c)
    s += part[((size_t)b * nchunks + c) * DEMB + i];
  u[idx] = s;
}

// --- 5: z = u @ Wv + bv   (WMMA, no guard: N=512 exact) ---------------------
__global__ void gemm_z_kernel(const float* __restrict__ u,
                              const float* __restrict__ Wv,
                              const float* __restrict__ bv,
                              float* __restrict__ z) {
  wmma_tile_f32<false, false>(u, DEMB, Wv, DEMB, bv, z, DEMB,
                              blockIdx.x * 16, blockIdx.y * 16, DEMB, DEMB);
}

// --- 6: out = z @ Wphi + bphi  (WMMA; only the tail tile takes the guarded
//         path, selected by a wave-uniform branch on blockIdx) ---------------
__global__ void gemm_out_kernel(const float* __restrict__ z,
                                const float* __restrict__ Wphi,
                                const float* __restrict__ bphi,
                                float* __restrict__ out, int C) {
  const int m0 = blockIdx.x * 16, n0 = blockIdx.y * 16;
  if (n0 + 16 <= C) {
    wmma_tile_f32<false, false>(z, DEMB, Wphi, C, bphi, out, C, m0, n0, DEMB, C);
  } else {
    wmma_tile_f32<false, true>(z, DEMB, Wphi, C, bphi, out, C, m0, n0, DEMB, C);
  }
}

extern "C" void kernel_launch(void* const* d_in, const int* in_sizes, int n_in,
                              void* d_out, int out_size, void* d_ws,
                              size_t ws_size, hipStream_t stream) {
  const float* x_items = (const float*)d_in[0];
  const float* x_query = (const float*)d_in[1];
  const float* Wq      = (const float*)d_in[2];
  const float* bq      = (const float*)d_in[3];
  const float* Wk      = (const float*)d_in[4];
  const float* bk      = (const float*)d_in[5];
  const float* Wv      = (const float*)d_in[6];
  const float* bv      = (const float*)d_in[7];
  const float* Wphi    = (const float*)d_in[8];
  const float* bphi    = (const float*)d_in[9];
  float* out = (float*)d_out;

  const int B = in_sizes[1];                        // 32
  const int N = in_sizes[0] / (B * DEMB);           // 4096
  const int C = in_sizes[9];                        // 1000
  const int nchunks = N / 256;                      // 16

  // workspace carve-up (floats); total ~1.9 MB
  float* w      = (float*)d_ws;
  float* Q      = w;                         // B*512
  float* wk     = Q + (size_t)B * DEMB;      // B*512
  float* qbk    = wk + (size_t)B * DEMB;     // B
  float* tau    = qbk + B;                   // B
  float* scores = tau + B;                   // B*N
  float* part   = scores + (size_t)B * N;    // B*nchunks*512
  float* u      = part + (size_t)B * nchunks * DEMB;  // B*512
  float* z      = u + (size_t)B * DEMB;      // B*512

  q_kernel<<<(B * DEMB + 255) / 256, 256, 0, stream>>>(x_query, Wq, bq, Q,
                                                       B * DEMB);
  qbk_kernel<<<1, B * 32, 0, stream>>>(Q, bk, qbk, B);
  gemm_wk_kernel<<<dim3(B / 16, DEMB / 16), 32, 0, stream>>>(Q, Wk, wk);
  score_kernel<<<(B * N) / 8, 256, 0, stream>>>(x_items, wk, qbk, scores, N);
  tau_kernel<<<B, 256, 0, stream>>>(scores, tau, N);
  upart_kernel<<<dim3(nchunks, B), 256, 0, stream>>>(x_items, scores, tau,
                                                     part, N, nchunks);
  ureduce_kernel<<<(B * DEMB + 255) / 256, 256, 0, stream>>>(part, u, nchunks,
                                                             B * DEMB);
  gemm_z_kernel<<<dim3(B / 16, DEMB / 16), 32, 0, stream>>>(u, Wv, bv, z);
  gemm_out_kernel<<<dim3(B / 16, (C + 15) / 16), 32, 0, stream>>>(z, Wphi,
                                                                  bphi, out, C);
}